// MBPert_52424370815395
// MI455X (gfx1250) — compile-verified
//
#include <hip/hip_runtime.h>

// MBPert GLV + fixed-step Dormand–Prince RK45, persistent-kernel formulation.
// Each workgroup owns 16 conditions (independent ODE columns) and runs all
// 100 steps locally. The 512x512 fp32 GEMM slab per RHS is done with
// V_WMMA_F32_16X16X4_F32 (CDNA5's fp32 matrix path). The stage-input matrix
// is double-buffered in LDS so only ONE workgroup barrier per RHS is needed.

typedef float v2f __attribute__((ext_vector_type(2)));
typedef float v8f __attribute__((ext_vector_type(8)));

constexpr int   NS    = 512;    // species
constexpr int   NC    = 1024;   // conditions
constexpr int   NCOL  = 16;     // conditions per workgroup
constexpr int   SRX   = 20;     // LDS row stride (dwords): bank-conflict free
constexpr int   NSTEP = 100;
constexpr float HSTEP = 0.2f;   // 20.0 / 100

// ---------------------------------------------------------------------------
// Pack A (row-major 512x512) into WMMA 16x4 fp32 A-fragment order so the hot
// loop streams contiguous 256B blocks per WMMA:
//   v2f index t = ((mt*128 + kt)*32 + lane), lane -> row mt*16+(lane&15),
//   cols kt*4 + 2*(lane>>4) .. +1   (ISA 16x4 f32 A layout)
// ---------------------------------------------------------------------------
__global__ void pack_A_kernel(const float* __restrict__ A,
                              float* __restrict__ Apack) {
  int t    = blockIdx.x * blockDim.x + threadIdx.x;   // 0 .. 512*512/2 - 1
  int lane = t & 31;
  int blk  = t >> 5;
  int kt   = blk & 127;
  int mt   = blk >> 7;
  int row  = mt * 16 + (lane & 15);
  int col  = kt * 4 + ((lane >> 4) << 1);
  const v2f* src = (const v2f*)(A + (size_t)row * NS + col);  // 8B aligned
  ((v2f*)Apack)[t] = *src;
}

// ---------------------------------------------------------------------------
// Persistent integrator: grid = 64 (NC/NCOL), block = 1024 (32 waves).
// Wave w computes rows [16w, 16w+16) x 16 conditions: one 16x16 accumulator.
// ---------------------------------------------------------------------------
__global__ void __launch_bounds__(1024)
mbpert_kernel(const float* __restrict__ x,
              const float* __restrict__ p,
              const float* __restrict__ r,
              const float* __restrict__ eps,
              const float* __restrict__ Apack,
              float* __restrict__ out) {
  // Double-buffered stage-input matrix: the only cross-wave data. 2 x 40KB.
  __shared__ float Xst[2 * NS * SRX];

  const int tid   = threadIdx.x;
  const int wave  = tid >> 5;
  const int lane  = tid & 31;
  const int nlane = lane & 15;      // N (condition within tile)
  const int hi    = lane >> 4;      // half-wave selector
  const int hi8   = hi << 3;
  const int hi2   = hi << 1;
  const int cond0 = blockIdx.x * NCOL;
  const int m0    = wave * 16;      // first species row of this wave

  // ---- load initial state (Fortran-stacked: X0[i,j] = x[j*NS + i]) --------
  float xcur[8];
  {
    const float* ip = x + (size_t)(cond0 + nlane) * NS + m0 + hi8;
    float4 a0 = *(const float4*)ip;
    float4 a1 = *(const float4*)(ip + 4);
    xcur[0] = a0.x; xcur[1] = a0.y; xcur[2] = a0.z; xcur[3] = a0.w;
    xcur[4] = a1.x; xcur[5] = a1.y; xcur[6] = a1.z; xcur[7] = a1.w;
  }

  // ---- per-lane invariant: rp[g] = r[m] + eps[m]*p[m, cond] ---------------
  // (C/D layout: acc[g] holds element (m0 + g + hi*8, nlane))
  float rp[8];
#pragma unroll
  for (int g = 0; g < 8; ++g) {
    int m = m0 + g + hi8;
    rp[g] = r[m] + eps[m] * p[(size_t)m * NC + cond0 + nlane];
  }

  // A-fragment stream base for this wave (contiguous 32KB, L2-resident).
  const v2f* __restrict__ Aw = (const v2f*)Apack + (size_t)wave * 4096; // 128*32

  float xstg[8];
  float k1[8], k2[8], k3[8], k4[8], k5[8], k6[8];

  // One RHS evaluation into buffer `buf`:
  //   publish xstg tile -> barrier -> 512-deep fp32 WMMA GEMM over Xst[buf]
  //   -> k = xstg * (rp + A@Xst).
  // Double buffering makes the post-GEMM barrier unnecessary: by the time any
  // wave writes buffer b for stage s+1 it has passed barrier(s), and every
  // wave signals barrier(s) only after completing its stage-(s-1) GEMM, the
  // last reader of buffer b.
  auto rhs_stage = [&](float* ko, int buf) {
    const int bo = buf * (NS * SRX);
#pragma unroll
    for (int g = 0; g < 8; ++g)
      Xst[bo + (m0 + g + hi8) * SRX + nlane] = xstg[g];
    __syncthreads();

    v8f acc = {0.f, 0.f, 0.f, 0.f, 0.f, 0.f, 0.f, 0.f};
#pragma unroll 8
    for (int kt = 0; kt < 128; ++kt) {
      v2f a = Aw[kt * 32 + lane];                 // coalesced global_load_b64
      int kb = (kt << 2) + hi2;                   // B frag rows kb, kb+1
      v2f b;
      b.x = Xst[bo + kb * SRX + nlane];
      b.y = Xst[bo + kb * SRX + SRX + nlane];
      acc = __builtin_amdgcn_wmma_f32_16x16x4_f32(
          /*neg_a=*/false, a, /*neg_b=*/false, b,
          /*c_mod=*/(short)0, acc, /*reuse_a=*/false, /*reuse_b=*/false);
    }
#pragma unroll
    for (int g = 0; g < 8; ++g)
      ko[g] = xstg[g] * (rp[g] + acc[g]);
  };

  // Dormand–Prince coefficients
  const float A21 = 0.2f;
  const float A31 = 3.0f / 40.0f,      A32 = 9.0f / 40.0f;
  const float A41 = 44.0f / 45.0f,     A42 = -56.0f / 15.0f,    A43 = 32.0f / 9.0f;
  const float A51 = 19372.0f / 6561.0f, A52 = -25360.0f / 2187.0f,
              A53 = 64448.0f / 6561.0f, A54 = -212.0f / 729.0f;
  const float A61 = 9017.0f / 3168.0f,  A62 = -355.0f / 33.0f,
              A63 = 46732.0f / 5247.0f, A64 = 49.0f / 176.0f,
              A65 = -5103.0f / 18656.0f;
  const float B1 = 35.0f / 384.0f, B3 = 500.0f / 1113.0f, B4 = 125.0f / 192.0f,
              B5 = -2187.0f / 6784.0f, B6 = 11.0f / 84.0f;

  for (int step = 0; step < NSTEP; ++step) {
#pragma unroll
    for (int g = 0; g < 8; ++g) xstg[g] = xcur[g];
    rhs_stage(k1, 0);

#pragma unroll
    for (int g = 0; g < 8; ++g)
      xstg[g] = xcur[g] + HSTEP * (A21 * k1[g]);
    rhs_stage(k2, 1);

#pragma unroll
    for (int g = 0; g < 8; ++g)
      xstg[g] = xcur[g] + HSTEP * (A31 * k1[g] + A32 * k2[g]);
    rhs_stage(k3, 0);

#pragma unroll
    for (int g = 0; g < 8; ++g)
      xstg[g] = xcur[g] + HSTEP * (A41 * k1[g] + A42 * k2[g] + A43 * k3[g]);
    rhs_stage(k4, 1);

#pragma unroll
    for (int g = 0; g < 8; ++g)
      xstg[g] = xcur[g] + HSTEP * (A51 * k1[g] + A52 * k2[g] + A53 * k3[g] +
                                   A54 * k4[g]);
    rhs_stage(k5, 0);

#pragma unroll
    for (int g = 0; g < 8; ++g)
      xstg[g] = xcur[g] + HSTEP * (A61 * k1[g] + A62 * k2[g] + A63 * k3[g] +
                                   A64 * k4[g] + A65 * k5[g]);
    rhs_stage(k6, 1);

#pragma unroll
    for (int g = 0; g < 8; ++g)
      xcur[g] += HSTEP * (B1 * k1[g] + B3 * k3[g] + B4 * k4[g] +
                          B5 * k5[g] + B6 * k6[g]);
  }

  // ---- store final state, Fortran-flattened: out[j*NS + i] ----------------
  {
    float* op = out + (size_t)(cond0 + nlane) * NS + m0 + hi8;
    float4 o0 = {xcur[0], xcur[1], xcur[2], xcur[3]};
    float4 o1 = {xcur[4], xcur[5], xcur[6], xcur[7]};
    *(float4*)op       = o0;
    *(float4*)(op + 4) = o1;
  }
}

// ---------------------------------------------------------------------------
extern "C" void kernel_launch(void* const* d_in, const int* in_sizes, int n_in,
                              void* d_out, int out_size, void* d_ws,
                              size_t ws_size, hipStream_t stream) {
  (void)in_sizes; (void)n_in; (void)out_size; (void)ws_size;
  const float* x   = (const float*)d_in[0];  // (NS*NC,) Fortran-stacked state
  const float* p   = (const float*)d_in[1];  // (NS, NC)
  const float* r   = (const float*)d_in[2];  // (NS,)
  const float* A   = (const float*)d_in[3];  // (NS, NS) row-major
  const float* eps = (const float*)d_in[4];  // (NS,)
  float* out   = (float*)d_out;
  float* Apack = (float*)d_ws;               // needs 1 MB scratch

  // One-time (per call) fragment-order repack of A: 512*512/2 v2f elements.
  pack_A_kernel<<<(NS * NS / 2 + 255) / 256, 256, 0, stream>>>(A, Apack);

  // 64 persistent workgroups x 32 waves; all 100 RK45 steps in-kernel.
  mbpert_kernel<<<NC / NCOL, 1024, 0, stream>>>(x, p, r, eps, Apack, out);
}